// OrdinalMixtureGCN_29549374996749
// MI455X (gfx1250) — compile-verified
//
#include <hip/hip_runtime.h>

typedef __attribute__((ext_vector_type(2))) float v2f;
typedef __attribute__((ext_vector_type(8))) float v8f;

#define N_U    100000
#define N_V    100000
#define DIM    64
#define R_SUP  5
#define E_EDGE 500000

// ---------------------------------------------------------------------------
// Kernel 1: cumulative (ordinal-mixture) weights.
// ws layout: [0, R*64*64)        -> Wu_cum
//            [R*64*64, 2*R*64*64) -> Wv_cum
// ---------------------------------------------------------------------------
__global__ void cumsum_weights(const float* __restrict__ wu,
                               const float* __restrict__ wv,
                               float* __restrict__ wcum) {
    int i = blockIdx.x * blockDim.x + threadIdx.x;
    const int DO = DIM * DIM;
    if (i >= 2 * DO) return;
    const float* src;
    float* dst;
    int j;
    if (i < DO) { src = wu; dst = wcum;              j = i;      }
    else        { src = wv; dst = wcum + R_SUP * DO; j = i - DO; }
    float acc = 0.f;
    #pragma unroll
    for (int r = 0; r < R_SUP; ++r) {
        acc += src[r * DO + j];
        dst[r * DO + j] = acc;
    }
}

// ---------------------------------------------------------------------------
// Kernel 2: fused (gather x-row) @ Wcum[r] -> scale -> atomic scatter.
// One wave32 handles 16 edges: A = 16x64 gathered rows, B = 64x64 W in LDS,
// D = A x B computed as 4 column tiles x 16 K-steps of v_wmma_f32_16x16x4_f32.
//
// f32 16x16x4 operand layout (ISA 7.12.2):
//   A (16x4, 2 VGPR): lanes 0-15 row M=lane  hold K=0,1 ; lanes 16-31 K=2,3
//   B (4x16, 2 VGPR): lanes 0-15 col N=lane  hold K=0,1 ; lanes 16-31 K=2,3
//   C/D (16x16, 8 VGPR): vgpr v, lane l -> row v+8*(l>>4), col l&15
// ---------------------------------------------------------------------------
__global__ __launch_bounds__(256)
void scatter_edges_wmma(const float* __restrict__ x_src,  // [N_src, 64]
                        const float* __restrict__ wcum,   // [R, 64, 64] (this side)
                        const int*   __restrict__ e_src,  // [R, E] gather idx into x_src
                        const int*   __restrict__ e_dst,  // [R, E] scatter idx into out
                        const float* __restrict__ e_val,  // [R, E]
                        float* __restrict__ out)          // [N_dst, 64] accumulator
{
    __shared__ float Wlds[DIM * DIM];   // 16 KB of the 320 KB WGP LDS

    const int r = blockIdx.y;
    // Stage this relation's 64x64 cumulated weight matrix into LDS (coalesced).
    const float* wsrc = wcum + r * DIM * DIM;
    for (int k = threadIdx.x; k < DIM * DIM; k += blockDim.x)
        Wlds[k] = wsrc[k];
    __syncthreads();

    const int wave = threadIdx.x >> 5;
    const int lane = threadIdx.x & 31;
    const int half = lane >> 4;      // 0: lanes 0-15, 1: lanes 16-31
    const int l16  = lane & 15;

    const int ebase = (blockIdx.x * 8 + wave) * 16;   // first edge of this wave
    if (ebase >= E_EDGE) return;     // E % 16 == 0 -> whole-wave uniform exit
    const int roff = r * E_EDGE;

    // Gather source row index for A-matrix row m = l16 (both halves need it).
    const int srow = e_src[roff + ebase + l16];

    // A fragments for all 16 K-steps: lane holds x[srow][4s + 2*half + {0,1}].
    const float* xrow = x_src + (size_t)srow * DIM + 2 * half;
    v2f afrag[16];
    #pragma unroll
    for (int s = 0; s < 16; ++s) {
        afrag[s].x = xrow[4 * s + 0];
        afrag[s].y = xrow[4 * s + 1];
    }

    // Per-lane edge scale + destination row for D rows m = v + 8*half.
    float scl[8];
    int   drow[8];
    #pragma unroll
    for (int v = 0; v < 8; ++v) {
        const int m = v + 8 * half;
        scl[v]  = e_val[roff + ebase + m];   // broadcast across 16 lanes
        drow[v] = e_dst[roff + ebase + m];
    }

    #pragma unroll
    for (int j = 0; j < 4; ++j) {            // output column tiles (O = 64)
        v8f acc = {};
        #pragma unroll
        for (int s = 0; s < 16; ++s) {       // K = 64 in steps of 4
            v2f b;
            b.x = Wlds[(4 * s + 2 * half + 0) * DIM + 16 * j + l16];
            b.y = Wlds[(4 * s + 2 * half + 1) * DIM + 16 * j + l16];
            acc = __builtin_amdgcn_wmma_f32_16x16x4_f32(
                      false, afrag[s], false, b, (short)0, acc, false, false);
        }
        // Scale per edge and scatter-add into the destination rows.
        #pragma unroll
        for (int v = 0; v < 8; ++v) {
            atomicAdd(&out[(size_t)drow[v] * DIM + 16 * j + l16],
                      acc[v] * scl[v]);
        }
    }
}

// ---------------------------------------------------------------------------
// Kernel 3: out = relu(out + bias)
// ---------------------------------------------------------------------------
__global__ void finalize_bias_relu(float* __restrict__ out,
                                   const float* __restrict__ bias_u,
                                   const float* __restrict__ bias_v) {
    const long long total = (long long)(N_U + N_V) * DIM;
    long long i = (long long)blockIdx.x * blockDim.x + threadIdx.x;
    if (i >= total) return;
    const int col = (int)(i & (DIM - 1));
    const float b = (i < (long long)N_U * DIM) ? bias_u[col] : bias_v[col];
    const float z = out[i] + b;
    out[i] = z > 0.f ? z : 0.f;
}

// ---------------------------------------------------------------------------
extern "C" void kernel_launch(void* const* d_in, const int* in_sizes, int n_in,
                              void* d_out, int out_size, void* d_ws, size_t ws_size,
                              hipStream_t stream) {
    const float* x_u    = (const float*)d_in[0];
    const float* x_v    = (const float*)d_in[1];
    const float* w_u    = (const float*)d_in[2];
    const float* w_v    = (const float*)d_in[3];
    const float* bias_u = (const float*)d_in[4];
    const float* bias_v = (const float*)d_in[5];
    const float* e_val  = (const float*)d_in[6];
    const int*   e_u    = (const int*)d_in[7];
    const int*   e_v    = (const int*)d_in[8];
    float* out  = (float*)d_out;
    float* wcum = (float*)d_ws;               // 2*R*64*64 floats = 160 KB

    // z accumulators start at zero (harness poisons d_out).
    hipMemsetAsync(d_out, 0, (size_t)(N_U + N_V) * DIM * sizeof(float), stream);

    cumsum_weights<<<dim3((2 * DIM * DIM + 255) / 256), 256, 0, stream>>>(w_u, w_v, wcum);

    const int waves_per_rel = E_EDGE / 16;                 // 31250
    dim3 grid((waves_per_rel + 7) / 8, R_SUP, 1);          // 8 waves / block

    // z_u: gather x_v rows via edge_v, transform by Wv_cum, scatter via edge_u.
    scatter_edges_wmma<<<grid, 256, 0, stream>>>(
        x_v, wcum + R_SUP * DIM * DIM, e_v, e_u, e_val, out);
    // z_v: gather x_u rows via edge_u, transform by Wu_cum, scatter via edge_v.
    scatter_edges_wmma<<<grid, 256, 0, stream>>>(
        x_u, wcum, e_u, e_v, e_val, out + (size_t)N_U * DIM);

    const long long total = (long long)(N_U + N_V) * DIM;
    finalize_bias_relu<<<(unsigned)((total + 255) / 256), 256, 0, stream>>>(
        out, bias_u, bias_v);
}